// Embedding_54305566490903
// MI455X (gfx1250) — compile-verified
//
#include <hip/hip_runtime.h>

// Embedding gather: out[row, :] = table[ids[row], :]
// table: [1,000,000 x 64] f32 (244 MB), ids: [819,200] int, out: [819,200 x 64] f32.
//
// Pure bandwidth kernel (~425 MB total traffic -> ~18 us at 23.3 TB/s).
// 16 lanes x 16B (global_load_b128 / global_store_b128) cover one 256-byte row;
// output + ids use non-temporal cache hints so the 192 MB L2 stays dedicated
// to the (nearly L2-resident, 244 MB) embedding table.

#define EMB_DIM 64          // floats per row
#define LANES_PER_ROW 16    // 16 lanes * 4 floats = 64 floats
#define BLOCK_THREADS 256   // 8 wave32 per block, 16 rows per block

// Native clang vector: legal operand for __builtin_nontemporal_{load,store}
// (HIP's float4 is a struct and is rejected by the builtin).
typedef __attribute__((ext_vector_type(4))) float v4f;

__global__ __launch_bounds__(BLOCK_THREADS)
void Embedding_54305566490903_kernel(const int* __restrict__ ids,
                                     const float* __restrict__ table,
                                     float* __restrict__ out,
                                     int num_rows) {
    const int gtid  = blockIdx.x * BLOCK_THREADS + threadIdx.x;
    const int row   = gtid >> 4;                    // which output row
    const int chunk = gtid & (LANES_PER_ROW - 1);   // which 16B piece of the row

    if (row >= num_rows) return;

    // ids are a read-once stream: non-temporal load (don't pollute L2).
    const int id = __builtin_nontemporal_load(ids + row);

    // Table read: regular temporal policy — we *want* the table resident in
    // the 192 MB L2; duplicate/hot ids hit L2 as long as the streaming output
    // doesn't evict it.
    const v4f* __restrict__ src =
        reinterpret_cast<const v4f*>(table + (size_t)id * EMB_DIM) + chunk;
    v4f v = *src;  // global_load_b128, coalesced 256B per half-wave

    // Output is a write-once stream: non-temporal store (TH_STORE_NT) so the
    // 210 MB output stream does not evict the table from L2.
    v4f* __restrict__ dst =
        reinterpret_cast<v4f*>(out + (size_t)row * EMB_DIM) + chunk;
    __builtin_nontemporal_store(v, dst);
}

extern "C" void kernel_launch(void* const* d_in, const int* in_sizes, int n_in,
                              void* d_out, int out_size, void* d_ws, size_t ws_size,
                              hipStream_t stream) {
    const int*   ids   = (const int*)d_in[0];     // setup_inputs order: ids first
    const float* table = (const float*)d_in[1];   // then table
    float*       out   = (float*)d_out;

    const int num_rows = in_sizes[0];             // 819,200

    const long long total_threads = (long long)num_rows * LANES_PER_ROW;
    const int grid = (int)((total_threads + BLOCK_THREADS - 1) / BLOCK_THREADS);

    Embedding_54305566490903_kernel<<<grid, BLOCK_THREADS, 0, stream>>>(
        ids, table, out, num_rows);
}